// LSTMLayer_71012989272504
// MI455X (gfx1250) — compile-verified
//
#include <hip/hip_runtime.h>
#include <math.h>

// Bidirectional LSTM, fused input-projection + recurrence, persistent kernel.
// B=16, T=2048, D=H=512, G=4H=2048.
//
// Grid: 64 persistent workgroups (32 per direction), 256 threads (8 wave32).
// Each WG owns 16 h-columns of ALL FOUR gates -> local cell update, one
// device-wide barrier per timestep per direction (needs >=64 free WGPs,
// trivially true on MI455X). Weight column slices (512x16 per wave, Wi for
// waves 0-3 / Wh for waves 4-7) live in LDS for the whole scan: 256KB of the
// 320KB CDNA5 WGP LDS (impossible on CDNA4's 64KB CUs).
//
// Matrix op: V_WMMA_F32_16X16X4_F32 (wave32, A=16x4 f32 in 2 VGPRs,
// B=4x16 f32 in 2 VGPRs, C/D=16x16 f32 in 8 VGPRs).
// h-state staging: GLOBAL_LOAD_ASYNC_TO_LDS_B128 (ASYNCcnt) at SCOPE_DEV,
// with a scalar-load fallback if the builtin is unavailable.

#define B_      16
#define T_      2048
#define D_      512
#define H_      512
#define G_      2048
#define NWG     32        // workgroups per direction
#define THREADS 256
#define WAVES   8

typedef __attribute__((ext_vector_type(2))) float v2f;
typedef __attribute__((ext_vector_type(8))) float v8f;

#if __has_builtin(__builtin_amdgcn_global_load_async_to_lds_b128)
#define HAVE_ASYNC_LDS 1
typedef __attribute__((ext_vector_type(4))) int v4i;
typedef __attribute__((address_space(1))) v4i gv4i;   // global int4
typedef __attribute__((address_space(3))) v4i lv4i;   // LDS int4
#else
#define HAVE_ASYNC_LDS 0
#endif

__device__ __forceinline__ void wait_asynccnt0() {
#if __has_builtin(__builtin_amdgcn_s_wait_asynccnt)
    __builtin_amdgcn_s_wait_asynccnt(0);
#else
    asm volatile("s_wait_asynccnt 0x0" ::: "memory");
#endif
}

// ---- dynamic LDS layout -----------------------------------------------------
// [0)                      weight slices: 8 waves * 512*16 f32, pair-interleaved
//                          idx = ((k>>1)*16 + n)*2 + (k&1)  -> ds_load_b64 of
//                          (k,n),(k+1,n); lanes 0..15 banks 2n..2n+1,
//                          lanes 16..31 banks +32 : conflict-free.
// [LDSW_FLOATS)            staged h state: 16 rows * 516 f32 (stride 516 ≡ 4
//                          mod 64 -> conflict-free 16-row b64 gathers; row
//                          stride 2064 B is 16B aligned for async b128 copies)
// [.. + AH_FLOATS)         gate tiles: 8 waves * 256 f32
#define LDSW_FLOATS (WAVES * 512 * 16)   // 65536 f32 = 256 KB
#define AH_STRIDE   516
#define AH_FLOATS   (16 * AH_STRIDE)     //  8256 f32 ≈ 32.25 KB
#define GATE_FLOATS (WAVES * 256)        //  2048 f32 = 8 KB
#define LDS_BYTES   ((LDSW_FLOATS + AH_FLOATS + GATE_FLOATS) * 4)  // 303360 B

// cpol for async loads: TH[2:0]=RT(0), SCOPE[4:3]=SCOPE_DEV(2) -> 2<<3
#define CPOL_SCOPE_DEV 16

__device__ __forceinline__ float sigmoidf_(float v) {
    return 1.0f / (1.0f + expf(-v));
}

__global__ void bilstm_init_kernel(unsigned* __restrict__ bar,
                                   float* __restrict__ hbuf) {
    int i = blockIdx.x * blockDim.x + threadIdx.x;
    if (i < 8) bar[i] = 0u;
    for (int j = i; j < 2 * B_ * H_; j += blockDim.x * gridDim.x)
        hbuf[j] = 0.0f;
}

__global__ void __launch_bounds__(THREADS, 1)
bilstm_kernel(const float* __restrict__ x,       // [B,T,D]
              const int*   __restrict__ lengths, // [B]
              const float* __restrict__ Wi_f, const float* __restrict__ Wh_f,
              const float* __restrict__ b_f,
              const float* __restrict__ Wi_b, const float* __restrict__ Wh_b,
              const float* __restrict__ b_b,
              float* __restrict__ out,           // [B,T,2H]
              unsigned* __restrict__ bar,        // 2 counters (per direction)
              float* __restrict__ hbuf_all)      // 2 * [B,H] h state
{
    extern __shared__ float lds[];
    float* ldsW  = lds;
    float* ldsAh = lds + LDSW_FLOATS;
    float* ldsG  = lds + LDSW_FLOATS + AH_FLOATS;

    const int tid   = threadIdx.x;
    const int wave  = tid >> 5;
    const int lane  = tid & 31;
    const int dir   = blockIdx.x / NWG;   // 0 = forward, 1 = backward
    const int hblk  = blockIdx.x % NWG;   // this WG's 16 h-columns
    const int gate  = wave & 3;           // i,f,g,o
    const int khalf = wave >> 2;          // 0: x@Wi half, 1: h@Wh half

    const float* Wi   = dir ? Wi_b : Wi_f;
    const float* Wh   = dir ? Wh_b : Wh_f;
    const float* bias = dir ? b_b  : b_f;
    float*    hbuf = hbuf_all + dir * (B_ * H_);
    unsigned* ctr  = bar + dir;

    // ---- one-time: stage this wave's 512x16 weight column slice into LDS ---
    {
        const float* Wsrc = (khalf ? Wh : Wi) + gate * H_ + hblk * 16;
        float* wdst = ldsW + wave * (512 * 16);
        for (int r0 = 0; r0 < 512; r0 += 2) {
            int r = r0 + (lane >> 4);
            int n = lane & 15;
            float v = Wsrc[(size_t)r * G_ + n];
            wdst[(((r >> 1) * 16 + n) << 1) | (r & 1)] = v;
        }
    }

    // per-lane A/B fragment geometry (16x16x4 f32 WMMA, wave32):
    //   lanes 0..15 : M/N = lane, K = k0 + {0,1}
    //   lanes 16..31: M/N = lane-16, K = k0 + {2,3}
    const int mrow = lane & 15;
    const int koff = (lane >> 4) * 2;
    const int len_m = lengths[mrow];

    // elementwise ownership: thread tid -> (batch em, column en)
    const int em = tid >> 4;
    const int en = tid & 15;
    const int hcol = hblk * 16 + en;
    const float bi = bias[0 * H_ + hcol];
    const float bf = bias[1 * H_ + hcol];
    const float bg = bias[2 * H_ + hcol];
    const float bo = bias[3 * H_ + hcol];
    const int elen = lengths[em];

    float c_state = 0.0f;
    __syncthreads();

    const float2* wlds2 = (const float2*)(ldsW + wave * (512 * 16));

    // staging geometry: thread -> 32 floats of the [16 x 512] h state
    const int sm  = tid >> 4;
    const int sc0 = (tid & 15) * 32;
    const float* ssrc = hbuf + sm * H_ + sc0;
    float*       sdst = ldsAh + sm * AH_STRIDE + sc0;

    for (int t = 0; t < T_; ++t) {
        // ---- stage full h state [16 x 512] from global into LDS ------------
#if HAVE_ASYNC_LDS
        {
#pragma unroll
            for (int j = 0; j < 8; ++j)
                __builtin_amdgcn_global_load_async_to_lds_b128(
                    (gv4i*)(ssrc + 4 * j), (lv4i*)(sdst + 4 * j),
                    0, CPOL_SCOPE_DEV);
            wait_asynccnt0();
        }
#else
        {
#pragma unroll
            for (int j = 0; j < 32; ++j)
                sdst[j] = __hip_atomic_load(ssrc + j, __ATOMIC_RELAXED,
                                            __HIP_MEMORY_SCOPE_AGENT);
        }
#endif
        __syncthreads();

        // ---- GEMM: 16x16 gate tile, K=512 via 128 x v_wmma_f32_16x16x4_f32 -
        v8f acc = {0.f, 0.f, 0.f, 0.f, 0.f, 0.f, 0.f, 0.f};
        if (khalf == 0) {
            // A rows = x[b, tin, :] (flip-indexed for the backward direction)
            int tin = dir ? ((T_ - 1 - t + len_m) & (T_ - 1)) : t;
            const float2* arow =
                (const float2*)(x + ((size_t)mrow * T_ + tin) * D_);
            {   // prefetch next step's row (global_prefetch_b8)
                int tnx = dir ? ((2 * T_ - 2 - t + len_m) & (T_ - 1))
                              : ((t + 1) & (T_ - 1));
                __builtin_prefetch(x + ((size_t)mrow * T_ + tnx) * D_, 0, 1);
            }
#pragma unroll 4
            for (int k0 = 0; k0 < 512; k0 += 4) {
                int kk = k0 + koff;
                float2 a2 = arow[kk >> 1];
                float2 b2 = wlds2[(kk >> 1) * 16 + mrow];
                v2f av = {a2.x, a2.y};
                v2f bv = {b2.x, b2.y};
                acc = __builtin_amdgcn_wmma_f32_16x16x4_f32(
                    false, av, false, bv, (short)0, acc, false, false);
            }
        } else {
            // A rows = h (previous step) from LDS
            const float2* ah2 = (const float2*)ldsAh;
#pragma unroll 4
            for (int k0 = 0; k0 < 512; k0 += 4) {
                int kk = k0 + koff;
                float2 a2 = ah2[(mrow * AH_STRIDE + kk) >> 1];
                float2 b2 = wlds2[(kk >> 1) * 16 + mrow];
                v2f av = {a2.x, a2.y};
                v2f bv = {b2.x, b2.y};
                acc = __builtin_amdgcn_wmma_f32_16x16x4_f32(
                    false, av, false, bv, (short)0, acc, false, false);
            }
        }

        // ---- spill gate tiles to LDS (C layout: VGPR r -> M=r / r+8) -------
        {
            float* g = ldsG + wave * 256;
#pragma unroll
            for (int r = 0; r < 8; ++r)
                g[(r + 8 * (lane >> 4)) * 16 + mrow] = acc[r];
        }
        __syncthreads();

        // ---- LSTM cell update: this thread owns (em, hcol) -----------------
        {
            int o = em * 16 + en;
            float iv = ldsG[0 * 256 + o] + ldsG[4 * 256 + o] + bi;
            float fv = ldsG[1 * 256 + o] + ldsG[5 * 256 + o] + bf;
            float gv = ldsG[2 * 256 + o] + ldsG[6 * 256 + o] + bg;
            float ov = ldsG[3 * 256 + o] + ldsG[7 * 256 + o] + bo;
            c_state = sigmoidf_(fv) * c_state + sigmoidf_(iv) * tanhf(gv);
            float h = sigmoidf_(ov) * tanhf(c_state);
            __hip_atomic_store(hbuf + em * H_ + hcol, h, __ATOMIC_RELAXED,
                               __HIP_MEMORY_SCOPE_AGENT);
            // fwd: out[b,t,hcol]; bwd: scatter through involutive flip index
            int tout = dir ? ((T_ - 1 - t + elen) & (T_ - 1)) : t;
            out[((size_t)em * T_ + tout) * (2 * H_) + dir * H_ + hcol] = h;
        }
        __syncthreads();

        // ---- device-wide barrier for this direction (32 WGs) ---------------
        if (tid == 0) {
            __threadfence();
            atomicAdd(ctr, 1u);
            unsigned target = (unsigned)(t + 1) * NWG;
            while (__hip_atomic_load(ctr, __ATOMIC_RELAXED,
                                     __HIP_MEMORY_SCOPE_AGENT) < target)
                __builtin_amdgcn_s_sleep(1);
            __threadfence();
        }
        __syncthreads();
    }
}

extern "C" void kernel_launch(void* const* d_in, const int* in_sizes, int n_in,
                              void* d_out, int out_size, void* d_ws,
                              size_t ws_size, hipStream_t stream) {
    (void)in_sizes; (void)n_in; (void)out_size; (void)ws_size;
    const float* x    = (const float*)d_in[0];
    const int*   lens = (const int*)d_in[1];
    const float* Wi_f = (const float*)d_in[2];
    const float* Wh_f = (const float*)d_in[3];
    const float* b_f  = (const float*)d_in[4];
    const float* Wi_b = (const float*)d_in[5];
    const float* Wh_b = (const float*)d_in[6];
    const float* b_b  = (const float*)d_in[7];
    float* out = (float*)d_out;

    unsigned* bar  = (unsigned*)d_ws;                    // 2 counters
    float*    hbuf = (float*)((char*)d_ws + 256);        // 2 * [B,H] f32

    // allow 303 KB dynamic LDS (CDNA5 WGP: 320 KB); harmless if a no-op
    (void)hipFuncSetAttribute((const void*)bilstm_kernel,
                              hipFuncAttributeMaxDynamicSharedMemorySize,
                              LDS_BYTES);

    bilstm_init_kernel<<<dim3(8), dim3(256), 0, stream>>>(bar, hbuf);
    bilstm_kernel<<<dim3(2 * NWG), dim3(THREADS), LDS_BYTES, stream>>>(
        x, lens, Wi_f, Wh_f, b_f, Wi_b, Wh_b, b_b, out, bar, hbuf);
}